// WAWL_60043642798466
// MI455X (gfx1250) — compile-verified
//
#include <hip/hip_runtime.h>

// WA wirelength (DREAMPlace-style), fp32, CSR segmented reduction.
// exp(x * 10) == exp2(x * 10 * log2(e)) ; INV_GAMMA = 10.0
#define INV_GAMMA_LOG2E 14.426950408889634f

// 4096 pins * 8B = 32KB LDS stage. Expected pins/block ~1024 (std ~32), so this
// is ~4x statistical headroom; WGP has 320KB LDS -> occupancy unaffected.
#define CAP_PINS 4096

#if defined(__has_builtin)
#if __has_builtin(__builtin_amdgcn_global_load_async_to_lds_b64)
#define HAVE_ASYNC_LDS_LOAD 1
#endif
#if __has_builtin(__builtin_amdgcn_s_wait_asynccnt)
#define HAVE_WAIT_ASYNC 1
#endif
#endif

// Builtin signature (from hipcc diagnostic): first param is
// 'v2i32 __device__*' (global addrspace), so: (global v2i* src, lds v2i* dst,
// imm offset, imm cpol).
typedef int v2i __attribute__((ext_vector_type(2)));
typedef __attribute__((address_space(1))) v2i as1_v2i;
typedef __attribute__((address_space(3))) v2i as3_v2i;

__global__ void wawl_zero_kernel(float* __restrict__ out) {
    if (threadIdx.x == 0 && blockIdx.x == 0) out[0] = 0.0f;
}

// Per-net WA wirelength for one coordinate pair; pins[0..n) holds (x,y).
// Called with either an LDS-backed pointer (ds_load path) or a global pointer;
// clang's infer-address-space specializes after inlining.
template <typename PinPtr>
static __device__ __forceinline__ float wa_net(PinPtr pins, int n) {
    float xmax = -3.402823466e38f, xmin = 3.402823466e38f;
    float ymax = -3.402823466e38f, ymin = 3.402823466e38f;
    for (int i = 0; i < n; ++i) {
        const float2 p = pins[i];
        xmax = fmaxf(xmax, p.x); xmin = fminf(xmin, p.x);
        ymax = fmaxf(ymax, p.y); ymin = fminf(ymin, p.y);
    }
    float spx = 0.f, sxpx = 0.f, snx = 0.f, sxnx = 0.f;
    float spy = 0.f, sxpy = 0.f, sny = 0.f, sxny = 0.f;
    for (int i = 0; i < n; ++i) {
        const float2 p = pins[i];
        const float epx = __builtin_amdgcn_exp2f((p.x - xmax) * INV_GAMMA_LOG2E);
        const float enx = __builtin_amdgcn_exp2f((xmin - p.x) * INV_GAMMA_LOG2E);
        const float epy = __builtin_amdgcn_exp2f((p.y - ymax) * INV_GAMMA_LOG2E);
        const float eny = __builtin_amdgcn_exp2f((ymin - p.y) * INV_GAMMA_LOG2E);
        spx += epx; sxpx += p.x * epx;
        snx += enx; sxnx += p.x * enx;
        spy += epy; sxpy += p.y * epy;
        sny += eny; sxny += p.y * eny;
    }
    // n >= 1 here -> each positive sum >= exp(0) = 1, divisions safe
    return (sxpx / spx - sxnx / snx) + (sxpy / spy - sxny / sny);
}

__global__ __launch_bounds__(256) void wawl_kernel(
    const float2* __restrict__ pos,             // [n_pins] (x,y) interleaved
    const float*  __restrict__ net_weights,     // [n_nets]
    const int*    __restrict__ netpin_start,    // [n_nets+1] CSR offsets
    const unsigned char* __restrict__ net_mask, // [n_nets] bool
    float* __restrict__ out,                    // [1]
    int n_nets)
{
    __shared__ float2 spins[CAP_PINS];

    const int net0   = blockIdx.x * 256;
    const int net    = net0 + threadIdx.x;
    const int netEnd = (net0 + 256 < n_nets) ? (net0 + 256) : n_nets;

    // Pins are sorted by net: this block's 256 nets own the contiguous pin
    // range [s0, e0). These are scalar (uniform) loads.
    const int s0  = netpin_start[net0];
    const int e0  = netpin_start[netEnd];
    const int cnt = e0 - s0;
    const bool useLds = (cnt <= CAP_PINS);  // uniform across the block

    if (useLds) {
#if defined(HAVE_ASYNC_LDS_LOAD)
        // CDNA5 async copy engine: global -> LDS without touching VGPR return
        // slots; tracked by ASYNCcnt.  One b64 (one pin) per lane per step.
        for (int i = threadIdx.x; i < cnt; i += 256) {
            __builtin_amdgcn_global_load_async_to_lds_b64(
                (as1_v2i*)(pos + s0 + i),
                (as3_v2i*)(spins + i),
                /*offset=*/0, /*cpol=*/0);
        }
#if defined(HAVE_WAIT_ASYNC)
        __builtin_amdgcn_s_wait_asynccnt(0);
#else
        asm volatile("s_wait_asynccnt 0x0" ::: "memory");
#endif
#else
        // Fallback stage: coalesced global_load_b64 + ds_store_b64.
        for (int i = threadIdx.x; i < cnt; i += 256)
            spins[i] = pos[s0 + i];
#endif
        __syncthreads();
    }

    float partial = 0.0f;
    if (net < n_nets) {
        const int s = netpin_start[net];
        const int e = netpin_start[net + 1];
        if (e > s && net_mask[net]) {
            const float wl = useLds ? wa_net(spins + (s - s0), e - s)   // ds_load_b64
                                    : wa_net(pos + s, e - s);           // global fallback
            partial = wl * net_weights[net];
        }
    }

    // ---- wave32 tree reduction (CDNA5 is wave32-only: 8 waves / 256 threads)
    for (int off = 16; off > 0; off >>= 1)
        partial += __shfl_down(partial, off, 32);

    __shared__ float smem[8];
    const int lane = threadIdx.x & 31;
    const int wave = threadIdx.x >> 5;
    if (lane == 0) smem[wave] = partial;
    __syncthreads();

    if (wave == 0) {
        float v = (lane < 8) ? smem[lane] : 0.0f;
        for (int off = 4; off > 0; off >>= 1)
            v += __shfl_down(v, off, 32);
        if (lane == 0) {
            // one f32 global atomic per block (8192 total) -> global_atomic_add_f32
            __hip_atomic_fetch_add(out, v, __ATOMIC_RELAXED, __HIP_MEMORY_SCOPE_AGENT);
        }
    }
}

extern "C" void kernel_launch(void* const* d_in, const int* in_sizes, int n_in,
                              void* d_out, int out_size, void* d_ws, size_t ws_size,
                              hipStream_t stream) {
    // setup_inputs() order:
    // 0: pos [N_PINS,2] f32   1: net_weights [N_NETS] f32
    // 2: flat_netpin (unused) 3: netpin_start [N_NETS+1] i32
    // 4: pin2net_map (unused) 5: net_mask [N_NETS] bool
    // 6: pin_mask (unused by reference)
    const float2* pos             = (const float2*)d_in[0];
    const float*  net_weights     = (const float*)d_in[1];
    const int*    netpin_start    = (const int*)d_in[3];
    const unsigned char* net_mask = (const unsigned char*)d_in[5];
    float* out = (float*)d_out;

    const int n_nets = in_sizes[1];

    // d_out is poisoned by the harness; zero it on-stream (graph-capture safe).
    wawl_zero_kernel<<<1, 32, 0, stream>>>(out);

    const int blocks = (n_nets + 255) / 256;
    wawl_kernel<<<blocks, 256, 0, stream>>>(pos, net_weights, netpin_start,
                                            net_mask, out, n_nets);
}